// InverInterpolator_8693013807472
// MI455X (gfx1250) — compile-verified
//
#include <hip/hip_runtime.h>
#include <hip/hip_bf16.h>

typedef __attribute__((ext_vector_type(2))) float v2f;
typedef __attribute__((ext_vector_type(4))) float f4;
typedef __attribute__((ext_vector_type(8))) float v8f;

#define BATCH 16
#define NROWS 32
#define IM_FE_INV (1.0f / 16.0f)

// ---------------------------------------------------------------------------
// Kernel 1: xy = Xg @ keypoint_g via V_WMMA_F32_16X16X4_F32, plus row mask.
// One wave (32 threads) per (batch, 16-row tile): grid = BATCH*2 waves.
// A tile 16x4 f32 (2 VGPRs): lanes 0-15 hold K={k0,k0+1}, lanes 16-31 K={k0+2,k0+3}.
// B tile 4x16 f32 (2 VGPRs): VGPR v, lane-half h -> row K = k0 + v + 2*h, col = lane%16.
// D 16x16 f32 (8 VGPRs): VGPR r, half h -> M = m0 + r + 8*h, N = lane%16.
// Only N=0 (x) and N=1 (y) columns are meaningful; cols 2..15 of B are zero.
// ---------------------------------------------------------------------------
__global__ void __launch_bounds__(32)
xy_mask_wmma(const float* __restrict__ Xg,   // (B, N, N)
             const float* __restrict__ kp,   // (B, N, 2)
             float* __restrict__ xy,         // ws: (B*N, 2) raw einsum result
             float* __restrict__ mask)       // ws: (B*N)
{
    const int wave = blockIdx.x;             // 0 .. 2*BATCH-1
    const int b    = wave >> 1;
    const int m0   = (wave & 1) << 4;        // 0 or 16
    const int lane = threadIdx.x & 31;
    const int half = lane >> 4;              // 0 | 1
    const int l15  = lane & 15;

    const float* Xgb = Xg + b * NROWS * NROWS;
    const float* kpb = kp + b * NROWS * 2;

    v8f c = {};
#pragma unroll
    for (int k0 = 0; k0 < NROWS; k0 += 4) {
        const int ka = k0 + half * 2;
        v2f a, bm;
        a.x  = Xgb[(m0 + l15) * NROWS + ka];
        a.y  = Xgb[(m0 + l15) * NROWS + ka + 1];
        bm.x = (l15 < 2) ? kpb[ka * 2 + l15]       : 0.0f;
        bm.y = (l15 < 2) ? kpb[(ka + 1) * 2 + l15] : 0.0f;
        c = __builtin_amdgcn_wmma_f32_16x16x4_f32(
                /*neg_a=*/false, a, /*neg_b=*/false, bm,
                /*c_mod=*/(short)0, c, /*reuse_a=*/false, /*reuse_b=*/false);
    }

    if (l15 < 2) {                           // lanes 0,1,16,17 carry cols N=0 (x) and N=1 (y)
        const int mbase = m0 + half * 8;
#pragma unroll
        for (int r = 0; r < 8; ++r)
            xy[(b * NROWS + mbase + r) * 2 + l15] = c[r];
    }

    if (lane < 16) {                         // mask[b, m0+lane] = sum_j Xg[b, m0+lane, j]
        float s = 0.0f;
        const float* row = Xgb + (m0 + lane) * NROWS;
#pragma unroll
        for (int j = 0; j < NROWS; ++j) s += row[j];
        mask[b * NROWS + m0 + lane] = s;
    }
}

// ---------------------------------------------------------------------------
// Kernel 2: zero-fill the 128 MB output with non-temporal b128 stores.
// This is the roofline-dominant part (~5.5 us at 23.3 TB/s). NT hint keeps the
// zeros from occupying L2 — they are write-once, mostly never re-read.
// ---------------------------------------------------------------------------
__global__ void __launch_bounds__(256)
zero_out(float* __restrict__ out, long n)
{
    const long n4     = n >> 2;
    f4* __restrict__ o4 = (f4*)out;
    const long stride = (long)gridDim.x * blockDim.x;
    long i            = (long)blockIdx.x * blockDim.x + threadIdx.x;
    const f4 z = {0.0f, 0.0f, 0.0f, 0.0f};
    for (long j = i; j < n4; j += stride)
        __builtin_nontemporal_store(z, o4 + j);
    if (i == 0)
        for (long j = n4 << 2; j < n; ++j) out[j] = 0.0f;
}

// ---------------------------------------------------------------------------
// Kernel 3: per (b,n) row, write the analytically-blurred 6x6 footprint:
//   out[p] = m*m * sum_{4 corners c} coef_c * G5[p - pos_c + 2]
// which equals gauss_blur(scatter_add(bilinear))*mask*mask, including
// coincident corners (clipping) and SAME zero padding (bounds checks).
// One block (64 thr) per row; threads 0..35 cover the 6x6 window.
// ---------------------------------------------------------------------------
__global__ void __launch_bounds__(64)
scatter_blur(const float* __restrict__ xy_ws,
             const float* __restrict__ mask_ws,
             const float* __restrict__ gk,     // (N,1,5,5); first 25 = the kernel
             float* __restrict__ out,
             const int* __restrict__ Hp,
             const int* __restrict__ Wp)
{
    const int row = blockIdx.x;               // b*N + n
    const int t   = threadIdx.x;
    const int H = *Hp, W = *Wp;

    const float m = mask_ws[row];
    const float x = xy_ws[row * 2 + 0] * IM_FE_INV - 0.5f;
    const float y = xy_ws[row * 2 + 1] * IM_FE_INV - 0.5f;

    const float maxx = (float)(W - 1), maxy = (float)(H - 1);
    const float lox = fminf(fmaxf(floorf(x), 0.0f), maxx);
    const float hix = fminf(fmaxf(ceilf(x),  0.0f), maxx);
    const float loy = fminf(fmaxf(floorf(y), 0.0f), maxy);
    const float hiy = fminf(fmaxf(ceilf(y),  0.0f), maxy);
    const float upx = x - lox, lwx = 1.0f - upx;
    const float upy = y - loy, lwy = 1.0f - upy;

    const int   ix[2] = {(int)lox, (int)hix};
    const int   iy[2] = {(int)loy, (int)hiy};
    const float wx[2] = {lwx, upx};
    const float wy[2] = {lwy, upy};

    if (t >= 36) return;
    const int dy = t / 6, dx = t % 6;
    const int py = iy[0] - 2 + dy;
    const int px = ix[0] - 2 + dx;
    if (py < 0 || py >= H || px < 0 || px >= W) return;

    float val = 0.0f;
#pragma unroll
    for (int cy = 0; cy < 2; ++cy) {
        const int oy = py - iy[cy] + 2;
        if (oy < 0 || oy > 4) continue;
#pragma unroll
        for (int cx = 0; cx < 2; ++cx) {
            const int ox = px - ix[cx] + 2;
            if (ox < 0 || ox > 4) continue;
            val += wy[cy] * wx[cx] * gk[oy * 5 + ox];
        }
    }
    out[((long)row * H + py) * W + px] = val * (m * m);
}

// ---------------------------------------------------------------------------
extern "C" void kernel_launch(void* const* d_in, const int* in_sizes, int n_in,
                              void* d_out, int out_size, void* d_ws, size_t ws_size,
                              hipStream_t stream)
{
    const float* Xg = (const float*)d_in[0];   // (16,32,32)
    const float* kp = (const float*)d_in[1];   // (16,32,2)
    const float* gk = (const float*)d_in[2];   // (32,1,5,5)
    const int*   Hp = (const int*)d_in[3];
    const int*   Wp = (const int*)d_in[4];
    float* out = (float*)d_out;

    float* xy   = (float*)d_ws;                // B*N*2 floats
    float* mask = xy + BATCH * NROWS * 2;      // B*N floats

    // 1) WMMA einsum + mask (32 single-wave blocks)
    xy_mask_wmma<<<BATCH * 2, 32, 0, stream>>>(Xg, kp, xy, mask);

    // 2) stream 128 MB of zeros (dominant, HBM-bound)
    zero_out<<<8192, 256, 0, stream>>>(out, (long)out_size);

    // 3) stamp 512 blurred 6x6 footprints
    scatter_blur<<<BATCH * NROWS, 64, 0, stream>>>(xy, mask, gk, out, Hp, Wp);
}